// CNFDynamicsGNN_50036368998564
// MI455X (gfx1250) — compile-verified
//
#include <hip/hip_runtime.h>
#include <hip/hip_bf16.h>

typedef __attribute__((ext_vector_type(2))) float v2f;
typedef __attribute__((ext_vector_type(8))) float v8f;

#define BN_EPS 1e-5f

// ---------------------------------------------------------------------------
// Degree / normalization precompute
// ---------------------------------------------------------------------------
__global__ void deg_accum_kernel(const int* __restrict__ dst,
                                 const float* __restrict__ ew,
                                 float* __restrict__ deg, int E) {
    int e = blockIdx.x * blockDim.x + threadIdx.x;
    if (e < E) atomicAdd(&deg[dst[e]], ew[e]);
}

__global__ void dinv_kernel(const float* __restrict__ deg,
                            float* __restrict__ dinv,
                            float* __restrict__ dinv2, int Nn) {
    int n = blockIdx.x * blockDim.x + threadIdx.x;
    if (n >= Nn) return;
    float d = deg[n] + 1.0f;                 // + self loop
    float di = (d > 0.0f) ? rsqrtf(d) : 0.0f;
    dinv[n] = di;
    dinv2[n] = di * di;
}

__global__ void norm_kernel(const int* __restrict__ src,
                            const int* __restrict__ dst,
                            const float* __restrict__ ew,
                            const float* __restrict__ dinv,
                            float* __restrict__ norm, int E) {
    int e = blockIdx.x * blockDim.x + threadIdx.x;
    if (e < E) norm[e] = dinv[src[e]] * ew[e] * dinv[dst[e]];
}

// ---------------------------------------------------------------------------
// Dense GEMM  Y[N, MT*16] = X[N,K] @ W[K, MT*16] via V_WMMA_F32_16X16X4_F32.
//
// One wave owns one 16-row tile and ALL MT column tiles (A fragment loaded
// once from global, reused MT times). W is staged in LDS pre-swizzled as
// K-pairs so each B fragment is a single ds_load_b64:
//   ldsW[p*M + c] = { W[2p][c], W[2p+1][c] }
// ---------------------------------------------------------------------------
template <int MT>
__global__ void gcn_gemm_wmma(const float* __restrict__ X,
                              const float* __restrict__ W,
                              float* __restrict__ Y,
                              int Nrows, int K) {
    constexpr int M = MT * 16;
    extern __shared__ float2 ldsW[];              // (K/2) x M float2
    const int tid = threadIdx.x;

    for (int i = tid; i < (K >> 1) * M; i += blockDim.x) {
        int p = i / M, c = i - p * M;
        ldsW[i] = make_float2(W[(2 * p) * M + c], W[(2 * p + 1) * M + c]);
    }
    __syncthreads();

    const int wave = tid >> 5;
    const int lane = tid & 31;
    const int rowTiles = (Nrows + 15) >> 4;
    const int tileRow = blockIdx.x * (blockDim.x >> 5) + wave;
    if (tileRow >= rowTiles) return;

    const int half = lane >> 4;   // 0: K subs 0/1, D rows 0..7; 1: K subs 2/3, D rows 8..15
    const int sub  = lane & 15;
    const int kb   = half * 2;

    int rowA = tileRow * 16 + sub;
    if (rowA >= Nrows) rowA = Nrows - 1;          // clamp loads; stores guarded below
    const float* __restrict__ xrow = X + (size_t)rowA * K;

    v8f acc[MT];
    #pragma unroll
    for (int t = 0; t < MT; ++t) acc[t] = v8f{};

    for (int kk = 0; kk < K; kk += 4) {
        float2 av = *(const float2*)(xrow + kk + kb);   // global_load_b64
        v2f a; a.x = av.x; a.y = av.y;
        const float2* __restrict__ brow = ldsW + ((kk >> 1) + half) * M + sub;
        #pragma unroll
        for (int t = 0; t < MT; ++t) {
            float2 bv = brow[t * 16];                   // ds_load_b64
            v2f b; b.x = bv.x; b.y = bv.y;
            acc[t] = __builtin_amdgcn_wmma_f32_16x16x4_f32(
                /*neg_a=*/false, a, /*neg_b=*/false, b,
                /*c_mod=*/(short)0, acc[t], /*reuse_a=*/false, /*reuse_b=*/false);
        }
    }

    #pragma unroll
    for (int r = 0; r < 8; ++r) {
        int rowD = tileRow * 16 + r + half * 8;
        if (rowD < Nrows) {
            float* __restrict__ yrow = Y + (size_t)rowD * M + sub;
            #pragma unroll
            for (int t = 0; t < MT; ++t) yrow[t * 16] = acc[t][r];
        }
    }
}

// ---------------------------------------------------------------------------
// AGG[n,:] = XW[n,:] * dinv2[n]   (self-loop message, also initializes AGG)
// ---------------------------------------------------------------------------
__global__ void agg_init_kernel(const float* __restrict__ XW,
                                const float* __restrict__ dinv2,
                                float* __restrict__ AGG, int Nrows, int M) {
    size_t i = (size_t)blockIdx.x * blockDim.x + threadIdx.x;   // float4 index
    size_t total = ((size_t)Nrows * M) >> 2;
    if (i >= total) return;
    int n = (int)(i / (size_t)(M >> 2));
    float4 x = ((const float4*)XW)[i];
    float s = dinv2[n];
    float4 o;
    o.x = x.x * s; o.y = x.y * s; o.z = x.z * s; o.w = x.w * s;
    ((float4*)AGG)[i] = o;
}

// ---------------------------------------------------------------------------
// Edge scatter: AGG[dst,:] += XW[src,:] * norm[e]; one wave32 per edge.
// ---------------------------------------------------------------------------
__global__ void scatter_kernel(const float* __restrict__ XW,
                               float* __restrict__ AGG,
                               const int* __restrict__ src,
                               const int* __restrict__ dst,
                               const float* __restrict__ norm,
                               int E, int M) {
    int wid  = (int)(((size_t)blockIdx.x * blockDim.x + threadIdx.x) >> 5);
    int lane = threadIdx.x & 31;
    if (wid >= E) return;
    int s = src[wid], d = dst[wid];
    float nv = norm[wid];
    const float4* __restrict__ xs = (const float4*)(XW + (size_t)s * M);
    float* __restrict__ ad = AGG + (size_t)d * M;
    for (int c4 = lane; c4 * 4 < M; c4 += 32) {
        float4 x = xs[c4];                                     // global_load_b128
        int c = c4 * 4;
        atomicAdd(&ad[c + 0], x.x * nv);
        atomicAdd(&ad[c + 1], x.y * nv);
        atomicAdd(&ad[c + 2], x.z * nv);
        atomicAdd(&ad[c + 3], x.w * nv);
    }
}

// ---------------------------------------------------------------------------
// Column sums / sum-of-squares of (AGG + b) for training-mode BatchNorm
// stats[0..M-1] = sum, stats[M..2M-1] = sumsq  (atomically accumulated)
// ---------------------------------------------------------------------------
__global__ void colstats_kernel(const float* __restrict__ AGG,
                                const float* __restrict__ b,
                                float* __restrict__ stats, int Nrows, int M) {
    int c = threadIdx.x % M;
    int rstep = blockDim.x / M;
    int r = blockIdx.x * rstep + (threadIdx.x / M);
    int stride = rstep * gridDim.x;
    float bc = b[c];
    float s = 0.0f, s2 = 0.0f;
    for (; r < Nrows; r += stride) {
        __builtin_prefetch(&AGG[(size_t)(r + stride) * M + c], 0, 1);
        float h = AGG[(size_t)r * M + c] + bc;
        s += h;
        s2 += h * h;
    }
    atomicAdd(&stats[c], s);
    atomicAdd(&stats[M + c], s2);
}

// ---------------------------------------------------------------------------
// OUT = relu( (AGG + b - mean) * rsqrt(var + eps) * gamma + beta )  (float4)
// ---------------------------------------------------------------------------
__global__ void bn_relu_kernel(const float* __restrict__ AGG,
                               const float* __restrict__ b,
                               const float* __restrict__ stats,
                               const float* __restrict__ gamma,
                               const float* __restrict__ beta,
                               float* __restrict__ OUT, int Nrows, int M) {
    size_t i = (size_t)blockIdx.x * blockDim.x + threadIdx.x;   // float4 index
    size_t total = ((size_t)Nrows * M) >> 2;
    if (i >= total) return;
    int c0 = (int)(i % (size_t)(M >> 2)) * 4;
    float invN = 1.0f / (float)Nrows;
    float4 hv = ((const float4*)AGG)[i];
    float h[4] = {hv.x, hv.y, hv.z, hv.w};
    float o[4];
    #pragma unroll
    for (int j = 0; j < 4; ++j) {
        int c = c0 + j;
        float m = stats[c] * invN;
        float v = stats[M + c] * invN - m * m;
        float y = (h[j] + b[c] - m) * rsqrtf(v + BN_EPS) * gamma[c] + beta[c];
        o[j] = fmaxf(y, 0.0f);
    }
    ((float4*)OUT)[i] = make_float4(o[0], o[1], o[2], o[3]);
}

// ---------------------------------------------------------------------------
// Final layer: OUT = AGG + b   (float4)
// ---------------------------------------------------------------------------
__global__ void add_bias_kernel(const float* __restrict__ AGG,
                                const float* __restrict__ b,
                                float* __restrict__ OUT, int Nrows, int M) {
    size_t i = (size_t)blockIdx.x * blockDim.x + threadIdx.x;   // float4 index
    size_t total = ((size_t)Nrows * M) >> 2;
    if (i >= total) return;
    int c0 = (int)(i % (size_t)(M >> 2)) * 4;
    float4 a = ((const float4*)AGG)[i];
    float4 o;
    o.x = a.x + b[c0 + 0];
    o.y = a.y + b[c0 + 1];
    o.z = a.z + b[c0 + 2];
    o.w = a.w + b[c0 + 3];
    ((float4*)OUT)[i] = o;
}

// ---------------------------------------------------------------------------
extern "C" void kernel_launch(void* const* d_in, const int* in_sizes, int n_in,
                              void* d_out, int out_size, void* d_ws, size_t ws_size,
                              hipStream_t stream) {
    const int D = 64, H = 128;
    const int N = in_sizes[0] / D;
    const int E = in_sizes[1];

    const float* z   = (const float*)d_in[0];
    const int*   src = (const int*)d_in[1];
    const int*   dst = (const int*)d_in[2];
    const float* ew  = (const float*)d_in[3];
    const float* W0  = (const float*)d_in[4];
    const float* b0  = (const float*)d_in[5];
    const float* W1  = (const float*)d_in[6];
    const float* b1  = (const float*)d_in[7];
    const float* W2  = (const float*)d_in[8];
    const float* b2  = (const float*)d_in[9];
    const float* g0  = (const float*)d_in[10];
    const float* be0 = (const float*)d_in[11];
    const float* g1  = (const float*)d_in[12];
    const float* be1 = (const float*)d_in[13];
    float* out = (float*)d_out;

    // Workspace layout (floats)
    float* ws      = (float*)d_ws;
    float* deg     = ws;                          // N
    float* dinv    = deg + N;                     // N
    float* dinv2   = dinv + N;                    // N
    float* normv   = dinv2 + N;                   // E
    float* stats   = normv + E;                   // 2*H
    float* bufXW   = stats + 2 * H;               // N*H
    float* bufAGG  = bufXW + (size_t)N * H;       // N*H
    float* bufH    = bufAGG + (size_t)N * H;      // N*H

    const int rowTiles = (N + 15) / 16;
    const int gemmBx = (rowTiles + 7) / 8;        // 8 waves per 256-thread block
    const int eThreads = (E * 32 + 255) / 256;    // one wave per edge

    // ---- degree / dinv / edge norm (shared by all 3 layers) ----
    hipMemsetAsync(deg, 0, (size_t)N * sizeof(float), stream);
    deg_accum_kernel<<<(E + 255) / 256, 256, 0, stream>>>(dst, ew, deg, E);
    dinv_kernel<<<(N + 255) / 256, 256, 0, stream>>>(deg, dinv, dinv2, N);
    norm_kernel<<<(E + 255) / 256, 256, 0, stream>>>(src, dst, ew, dinv, normv, E);

    // ================= Layer 1: [N,64] -> [N,128], BN+ReLU =================
    gcn_gemm_wmma<8><<<gemmBx, 256, D * H * sizeof(float), stream>>>(
        z, W0, bufXW, N, D);
    agg_init_kernel<<<(int)(((size_t)N * H / 4 + 255) / 256), 256, 0, stream>>>(
        bufXW, dinv2, bufAGG, N, H);
    scatter_kernel<<<eThreads, 256, 0, stream>>>(bufXW, bufAGG, src, dst, normv, E, H);
    hipMemsetAsync(stats, 0, 2 * H * sizeof(float), stream);
    colstats_kernel<<<256, 256, 0, stream>>>(bufAGG, b0, stats, N, H);
    bn_relu_kernel<<<(int)(((size_t)N * H / 4 + 255) / 256), 256, 0, stream>>>(
        bufAGG, b0, stats, g0, be0, bufH, N, H);

    // ================= Layer 2: [N,128] -> [N,128], BN+ReLU ================
    gcn_gemm_wmma<8><<<gemmBx, 256, H * H * sizeof(float), stream>>>(
        bufH, W1, bufXW, N, H);
    agg_init_kernel<<<(int)(((size_t)N * H / 4 + 255) / 256), 256, 0, stream>>>(
        bufXW, dinv2, bufAGG, N, H);
    scatter_kernel<<<eThreads, 256, 0, stream>>>(bufXW, bufAGG, src, dst, normv, E, H);
    hipMemsetAsync(stats, 0, 2 * H * sizeof(float), stream);
    colstats_kernel<<<256, 256, 0, stream>>>(bufAGG, b1, stats, N, H);
    bn_relu_kernel<<<(int)(((size_t)N * H / 4 + 255) / 256), 256, 0, stream>>>(
        bufAGG, b1, stats, g1, be1, bufH, N, H);

    // ================= Layer 3: [N,128] -> [N,64], bias only ===============
    gcn_gemm_wmma<4><<<gemmBx, 256, H * D * sizeof(float), stream>>>(
        bufH, W2, bufXW, N, H);
    agg_init_kernel<<<(int)(((size_t)N * D / 4 + 255) / 256), 256, 0, stream>>>(
        bufXW, dinv2, bufAGG, N, D);
    scatter_kernel<<<eThreads, 256, 0, stream>>>(bufXW, bufAGG, src, dst, normv, E, D);
    add_bias_kernel<<<(int)(((size_t)N * D / 4 + 255) / 256), 256, 0, stream>>>(
        bufAGG, b2, out, N, D);
}